// NeuralRoughVolEngine_32427003084858
// MI455X (gfx1250) — compile-verified
//
#include <hip/hip_runtime.h>
#include <hip/hip_bf16.h>
#include <math.h>

typedef __attribute__((ext_vector_type(2))) float v2f;
typedef __attribute__((ext_vector_type(8))) float v8f;
typedef __attribute__((ext_vector_type(4))) unsigned int u32x4;
typedef __attribute__((ext_vector_type(8))) int i32x8;
typedef __attribute__((ext_vector_type(4))) int i32x4;

#define NT 365
#define TPAD 368          // padded K dimension
#define LTW 384           // Lt row width (12 * 32 columns)
#define CT 12             // 32-column tiles
#define ROWS 128          // batch rows per GEMM block
#define SPAD 372          // LDS row stride for z tile (bank-conflict free)
#define TRI 66795         // 365*366/2 packed lower triangle
#define PS 64             // samples per param-net block
#define HD 128
#define HPAD 132

#if defined(__has_builtin)
# if __has_builtin(__builtin_amdgcn_tensor_load_to_lds)
#  define HAVE_TDM 1
# endif
#endif
#ifndef HAVE_TDM
# define HAVE_TDM 0
#endif

// ---------------------------------------------------------------- helpers
__device__ inline float gelu_exact(float x) {
    return 0.5f * x * (1.0f + erff(x * 0.70710678118654752f));
}
__device__ inline float softplus_f(float x) {
    return fmaxf(x, 0.0f) + log1pf(expf(-fabsf(x)));
}
__device__ inline int tri_row(int p) {
    int r = (int)((sqrtf(8.0f * (float)p + 1.0f) - 1.0f) * 0.5f);
    while ((r + 1) * (r + 2) / 2 <= p) ++r;
    while (r * (r + 1) / 2 > p) --r;
    return r;
}

#if HAVE_TDM
// Simple 2D tile load: dense LDS destination, strided global rows.
// Units are elements of 4 bytes. tensor dims == tile dims (no OOB clipping).
__device__ inline void tdm_load_2d(unsigned lds_off, const float* gaddr,
                                   unsigned tile_d0, unsigned tile_d1,
                                   unsigned stride_d0)
{
    unsigned long long ga = (unsigned long long)(uintptr_t)gaddr;
    u32x4 g0 = { 1u,                                   // count=1 (valid user D#)
                 lds_off,                              // lds_addr
                 (unsigned)(ga & 0xFFFFFFFFu),
                 (unsigned)((ga >> 32) & 0x01FFFFFFu) | (2u << 30) };  // type=2
    i32x8 g1 = { (int)(2u << 16),                      // data_size = 4B
                 (int)((tile_d0 & 0xFFFFu) << 16),     // tensor_dim0[15:0]
                 (int)((tile_d0 >> 16) | ((tile_d1 & 0xFFFFu) << 16)),
                 (int)((tile_d1 >> 16) | (tile_d0 << 16)),  // tile_dim0
                 (int)(tile_d1 & 0xFFFFu),             // tile_dim1 (tile_dim2=0)
                 (int)stride_d0,                       // tensor_dim0_stride lo
                 0, 0 };
    i32x4 gz = { 0, 0, 0, 0 };
    i32x8 gz8 = { 0, 0, 0, 0, 0, 0, 0, 0 };
    __builtin_amdgcn_tensor_load_to_lds(g0, g1, gz, gz, gz8, 0);
}

// Iterated row loads: niter rows of rowlen elements, global step grow elements,
// LDS step lrow elements (expresses a padded LDS row stride).
__device__ inline void tdm_load_rows(unsigned lds_off, const float* gaddr,
                                     unsigned rowlen, unsigned niter,
                                     unsigned grow, unsigned lrow)
{
    unsigned long long ga = (unsigned long long)(uintptr_t)gaddr;
    u32x4 g0 = { 1u, lds_off,
                 (unsigned)(ga & 0xFFFFFFFFu),
                 (unsigned)((ga >> 32) & 0x01FFFFFFu) | (2u << 30) };
    i32x8 g1 = { (int)((2u << 16) | (1u << 19)),       // data_size=4B, iterate
                 (int)((rowlen & 0xFFFFu) << 16),      // tensor_dim0
                 (int)((rowlen >> 16) | (1u << 16)),   // tensor_dim1 = 1
                 (int)(rowlen << 16),                  // tile_dim0 = rowlen
                 (int)1,                               // tile_dim1 = 1
                 (int)grow,                            // stride (unused for 1 row)
                 0, 0 };
    i32x4 g2 = { 0,                                    // tensor_dim2
                 (int)lrow,                            // lds_addr_increment
                 (int)grow,                            // global_addr_increment lo
                 (int)((niter - 1) << 16) };           // iterate_count
    i32x4 g3 = { 0, 0, 0, 0 };
    i32x8 gz8 = { 0, 0, 0, 0, 0, 0, 0, 0 };
    __builtin_amdgcn_tensor_load_to_lds(g0, g1, g2, g3, gz8, 0);
}

__device__ inline unsigned lds_off_of(const void* p) {
    return (unsigned)(uintptr_t)p;      // low 32 bits of flat shared address
}
#endif

// =====================================================================
// Kernel 1: param net, 64 samples / block, 256 threads (8 waves)
// =====================================================================
__global__ __launch_bounds__(256) void param_kernel(
    const float* __restrict__ pv, const float* __restrict__ cond,
    const float* __restrict__ W1, const float* __restrict__ b1,
    const float* __restrict__ g1, const float* __restrict__ be1,
    const float* __restrict__ W2, const float* __restrict__ b2,
    const float* __restrict__ g2, const float* __restrict__ be2,
    const float* __restrict__ W3, const float* __restrict__ b3,
    float* __restrict__ logpv, float* __restrict__ xiA,
    float* __restrict__ hxi2, float* __restrict__ scaleA)
{
    __shared__ float cs[PS][10];
    __shared__ float hbuf[PS][HPAD];
    __shared__ float ws2[HD * HD];
    __shared__ float red1[PS * 4];
    __shared__ float red2[PS * 4];
    __shared__ float mv[PS][2];
    __shared__ float pout[PS][4];

    const int tid = threadIdx.x;
    const int b0 = blockIdx.x * PS;

    for (int i = tid; i < PS * 10; i += 256) cs[i / 10][i % 10] = cond[(size_t)b0 * 10 + i];
    for (int i = tid; i < HD * HD; i += 256) {
        int k = i >> 7, n = i & 127;
        ws2[(k >> 1) * 256 + n * 2 + (k & 1)] = W2[i];
    }
    __syncthreads();

    const int s = tid >> 2;
    const int q = tid & 3;
    {
        float sum = 0.0f, ssq = 0.0f;
        for (int f = q * 32; f < q * 32 + 32; ++f) {
            float acc = b1[f];
#pragma unroll
            for (int c = 0; c < 10; ++c) acc = fmaf(cs[s][c], W1[c * 128 + f], acc);
            float x = gelu_exact(acc);
            hbuf[s][f] = x;
            sum += x; ssq += x * x;
        }
        red1[tid] = sum; red2[tid] = ssq;
    }
    __syncthreads();
    if (q == 0) {
        float sum = red1[tid] + red1[tid + 1] + red1[tid + 2] + red1[tid + 3];
        float ssq = red2[tid] + red2[tid + 1] + red2[tid + 2] + red2[tid + 3];
        float m = sum * (1.0f / 128.0f);
        float v = ssq * (1.0f / 128.0f) - m * m;
        mv[s][0] = m; mv[s][1] = rsqrtf(v + 1e-5f);
    }
    __syncthreads();
    {
        float m = mv[s][0], r = mv[s][1];
        for (int f = q * 32; f < q * 32 + 32; ++f)
            hbuf[s][f] = (hbuf[s][f] - m) * r * g1[f] + be1[f];
    }
    __syncthreads();

    const int wave = tid >> 5, lane = tid & 31;
    const int half = lane >> 4, l15 = lane & 15;
    const int mt = (wave & 3) * 16;
    const int nb = (wave >> 2) * 64;

    v8f a0 = {}, a1 = {}, a2 = {}, a3 = {};
    for (int k0 = 0; k0 < HD; k0 += 4) {
        const int kb = k0 + 2 * half;
        v2f a = *(const v2f*)&hbuf[mt + l15][kb];
        const float* bp = &ws2[(kb >> 1) * 256];
        v2f q0 = *(const v2f*)&bp[(nb +      l15) * 2];
        v2f q1 = *(const v2f*)&bp[(nb + 16 + l15) * 2];
        v2f q2 = *(const v2f*)&bp[(nb + 32 + l15) * 2];
        v2f q3 = *(const v2f*)&bp[(nb + 48 + l15) * 2];
        a0 = __builtin_amdgcn_wmma_f32_16x16x4_f32(false, a, false, q0, (short)0, a0, false, false);
        a1 = __builtin_amdgcn_wmma_f32_16x16x4_f32(false, a, false, q1, (short)0, a1, false, false);
        a2 = __builtin_amdgcn_wmma_f32_16x16x4_f32(false, a, false, q2, (short)0, a2, false, false);
        a3 = __builtin_amdgcn_wmma_f32_16x16x4_f32(false, a, false, q3, (short)0, a3, false, false);
    }
    __syncthreads();

    {
        const float bb0 = b2[nb + l15];
        const float bb1 = b2[nb + 16 + l15];
        const float bb2 = b2[nb + 32 + l15];
        const float bb3 = b2[nb + 48 + l15];
#pragma unroll
        for (int v = 0; v < 8; ++v) {
            const int m = mt + v + 8 * half;
            hbuf[m][nb +      l15] = gelu_exact(a0[v] + bb0);
            hbuf[m][nb + 16 + l15] = gelu_exact(a1[v] + bb1);
            hbuf[m][nb + 32 + l15] = gelu_exact(a2[v] + bb2);
            hbuf[m][nb + 48 + l15] = gelu_exact(a3[v] + bb3);
        }
    }
    __syncthreads();

    {
        float sum = 0.0f, ssq = 0.0f;
        for (int f = q * 32; f < q * 32 + 32; ++f) {
            float x = hbuf[s][f];
            sum += x; ssq += x * x;
        }
        red1[tid] = sum; red2[tid] = ssq;
    }
    __syncthreads();
    if (q == 0) {
        float sum = red1[tid] + red1[tid + 1] + red1[tid + 2] + red1[tid + 3];
        float ssq = red2[tid] + red2[tid + 1] + red2[tid + 2] + red2[tid + 3];
        float m = sum * (1.0f / 128.0f);
        float v = ssq * (1.0f / 128.0f) - m * m;
        mv[s][0] = m; mv[s][1] = rsqrtf(v + 1e-5f);
    }
    __syncthreads();
    {
        float m = mv[s][0], r = mv[s][1];
        for (int f = q * 32; f < q * 32 + 32; ++f)
            hbuf[s][f] = (hbuf[s][f] - m) * r * g2[f] + be2[f];
    }
    __syncthreads();

    if (tid < PS * 3) {
        int ss = tid / 3, j = tid - ss * 3;
        float p = b3[j];
        for (int c = 0; c < HD; ++c) p = fmaf(hbuf[ss][c], W3[c * 3 + j], p);
        pout[ss][j] = p;
    }
    __syncthreads();
    if (tid < PS) {
        float xi = softplus_f(pout[tid][0]);
        float fv = softplus_f(pout[tid][2]);
        float p0 = pv[b0 + tid];
        logpv[b0 + tid]  = logf(p0);
        xiA[b0 + tid]    = xi;
        hxi2[b0 + tid]   = 0.5f * xi * xi;
        scaleA[b0 + tid] = sqrtf(fv / p0);
    }
}

// =====================================================================
// Kernel 2: fractional covariance + in-LDS packed Cholesky + write L^T
// =====================================================================
__global__ __launch_bounds__(512) void chol_kernel(float* __restrict__ t2h,
                                                   float* __restrict__ Lt)
{
    __shared__ float T2[TPAD];
    __shared__ float D2[NT];
    __shared__ float A[TRI];

    const int tid = threadIdx.x;

    for (int i = tid; i < TPAD; i += 512) {
        float val = 0.0f;
        if (i < NT) val = powf((float)i * (1.0f / 364.0f), 0.2f);
        T2[i] = val;
        t2h[i] = val;
    }
    for (int i = tid; i < NT; i += 512)
        D2[i] = powf((float)i * (1.0f / 364.0f) + 1e-10f, 0.2f);
    __syncthreads();

    for (int p = tid; p < TRI; p += 512) {
        int i = tri_row(p);
        int k = p - i * (i + 1) / 2;
        float val = 0.5f * (T2[i] + T2[k] - D2[i - k]);
        if (i == k) val += 1e-6f;
        A[p] = val;
    }
    __syncthreads();

    for (int j = 0; j < NT; ++j) {
        const int dj = j * (j + 1) / 2 + j;
        if (tid == 0) A[dj] = sqrtf(A[dj]);
        __syncthreads();
        const float inv = 1.0f / A[dj];
        for (int i = j + 1 + tid; i < NT; i += 512)
            A[i * (i + 1) / 2 + j] *= inv;
        __syncthreads();
        const int mrem = NT - 1 - j;
        const int M = mrem * (mrem + 1) / 2;
        for (int p = tid; p < M; p += 512) {
            int r = tri_row(p);
            int c = p - r * (r + 1) / 2;
            int i = j + 1 + r;
            int k = j + 1 + c;
            A[i * (i + 1) / 2 + k] -= A[i * (i + 1) / 2 + j] * A[k * (k + 1) / 2 + j];
        }
        __syncthreads();
    }

    for (int idx = tid; idx < TPAD * LTW; idx += 512) {
        int k = idx / LTW, i = idx - k * LTW;
        float v = 0.0f;
        if (i < NT && k < NT && k <= i) v = A[i * (i + 1) / 2 + k];
        Lt[idx] = v;
    }
}

// =====================================================================
// Kernel 3: fBm = z @ L^T via WMMA; TDM-staged, double-buffered B tiles;
// 32-col tiles -> 2 independent WMMA chains per wave; fused epilogue
// =====================================================================
__global__ __launch_bounds__(256) void gemm_kernel(
    const float* __restrict__ z, const float* __restrict__ Lt,
    const float* __restrict__ t2h,
    const float* __restrict__ logpv, const float* __restrict__ xiA,
    const float* __restrict__ hxi2, const float* __restrict__ scaleA,
    float* __restrict__ out)
{
    __shared__ float zs[ROWS][SPAD];       // 190464 B
    __shared__ float bsA[2][TPAD * 16];    // cols nt*32 + 0..15  (dense [k][16])
    __shared__ float bsB[2][TPAD * 16];    // cols nt*32 + 16..31
    __shared__ float prm[4][ROWS];
    __shared__ float t2s[TPAD];

    const int tid = threadIdx.x;
    const int row0 = blockIdx.x * ROWS;
    const int wave = tid >> 5;
    const int lane = tid & 31;
    const int half = lane >> 4;
    const int l15  = lane & 15;
    const int mrow = wave * 16;

#if HAVE_TDM
    if (wave == 0) {
        tdm_load_rows(lds_off_of(&zs[0][0]), z + (size_t)row0 * NT,
                      NT, ROWS, NT, SPAD);
        tdm_load_2d(lds_off_of(&bsA[0][0]), Lt +  0, 16, TPAD, LTW);
        tdm_load_2d(lds_off_of(&bsB[0][0]), Lt + 16, 16, TPAD, LTW);
    }
    // zero the K-pad columns TDM does not write (k = 365..367)
    for (int idx = tid; idx < ROWS * 3; idx += 256) {
        int r = idx / 3, c = NT + idx - r * 3;
        zs[r][c] = 0.0f;
    }
#else
    for (int idx = tid; idx < ROWS * TPAD; idx += 256) {
        int r = idx / TPAD, c = idx - r * TPAD;
        zs[r][c] = (c < NT) ? z[(size_t)(row0 + r) * NT + c] : 0.0f;
    }
#pragma unroll
    for (int i = 0; i < 23; ++i) {
        int idx = tid + i * 256;
        bsA[0][idx] = Lt[(size_t)(idx >> 4) * LTW + (idx & 15)];
        bsB[0][idx] = Lt[(size_t)(idx >> 4) * LTW + 16 + (idx & 15)];
    }
#endif
    for (int idx = tid; idx < ROWS; idx += 256) {
        prm[0][idx] = logpv[row0 + idx];
        prm[1][idx] = xiA[row0 + idx];
        prm[2][idx] = hxi2[row0 + idx];
        prm[3][idx] = scaleA[row0 + idx];
    }
    for (int idx = tid; idx < TPAD; idx += 256) t2s[idx] = t2h[idx];
#if HAVE_TDM
    if (wave == 0) __builtin_amdgcn_s_wait_tensorcnt(0);
#endif
    __syncthreads();

    int cur = 0;
    for (int nt = 0; nt < CT; ++nt) {
#if HAVE_TDM
        if (wave == 0 && nt + 1 < CT) {
            tdm_load_2d(lds_off_of(&bsA[cur ^ 1][0]), Lt + (nt + 1) * 32,      16, TPAD, LTW);
            tdm_load_2d(lds_off_of(&bsB[cur ^ 1][0]), Lt + (nt + 1) * 32 + 16, 16, TPAD, LTW);
        }
#else
        float preA[23], preB[23];
        if (nt + 1 < CT) {
#pragma unroll
            for (int i = 0; i < 23; ++i) {
                int idx = tid + i * 256;
                preA[i] = Lt[(size_t)(idx >> 4) * LTW + (nt + 1) * 32 + (idx & 15)];
                preB[i] = Lt[(size_t)(idx >> 4) * LTW + (nt + 1) * 32 + 16 + (idx & 15)];
            }
        }
#endif
        const float* bA = bsA[cur];
        const float* bB = bsB[cur];
        v8f acc0 = {}, acc1 = {};
#pragma unroll 4
        for (int k0 = 0; k0 < TPAD; k0 += 4) {
            const int kb = k0 + 2 * half;
            v2f a = *(const v2f*)&zs[mrow + l15][kb];
            v2f b0; b0.x = bA[kb * 16 + l15]; b0.y = bA[(kb + 1) * 16 + l15];
            v2f b1; b1.x = bB[kb * 16 + l15]; b1.y = bB[(kb + 1) * 16 + l15];
            acc0 = __builtin_amdgcn_wmma_f32_16x16x4_f32(false, a, false, b0, (short)0, acc0, false, false);
            acc1 = __builtin_amdgcn_wmma_f32_16x16x4_f32(false, a, false, b1, (short)0, acc1, false, false);
        }

#if !HAVE_TDM
        if (nt + 1 < CT) {
#pragma unroll
            for (int i = 0; i < 23; ++i) {
                int idx = tid + i * 256;
                bsA[cur ^ 1][idx] = preA[i];
                bsB[cur ^ 1][idx] = preB[i];
            }
        }
#endif
        // fused epilogue: vol = exp(logpv + xi*fBm - 0.5*xi^2 * t^2H) * scale
        const int col0 = nt * 32 + l15;
        if (col0 < NT) {
            const float tc = t2s[col0];
#pragma unroll
            for (int v = 0; v < 8; ++v) {
                const int rl = mrow + v + 8 * half;
                float lv = prm[0][rl] + prm[1][rl] * acc0[v] - prm[2][rl] * tc;
                out[(size_t)(row0 + rl) * NT + col0] = expf(lv) * prm[3][rl];
            }
        }
        const int col1 = nt * 32 + 16 + l15;
        if (col1 < NT) {
            const float tc = t2s[col1];
#pragma unroll
            for (int v = 0; v < 8; ++v) {
                const int rl = mrow + v + 8 * half;
                float lv = prm[0][rl] + prm[1][rl] * acc1[v] - prm[2][rl] * tc;
                out[(size_t)(row0 + rl) * NT + col1] = expf(lv) * prm[3][rl];
            }
        }
#if HAVE_TDM
        if (wave == 0) __builtin_amdgcn_s_wait_tensorcnt(0);
#endif
        __syncthreads();
        cur ^= 1;
    }
}

// =====================================================================
extern "C" void kernel_launch(void* const* d_in, const int* in_sizes, int n_in,
                              void* d_out, int out_size, void* d_ws, size_t ws_size,
                              hipStream_t stream) {
    (void)n_in; (void)out_size; (void)ws_size;
    const float* pv   = (const float*)d_in[0];
    const float* cond = (const float*)d_in[1];
    const float* z    = (const float*)d_in[2];
    const float* W1   = (const float*)d_in[3];
    const float* b1   = (const float*)d_in[4];
    const float* g1   = (const float*)d_in[5];
    const float* be1  = (const float*)d_in[6];
    const float* W2   = (const float*)d_in[7];
    const float* b2   = (const float*)d_in[8];
    const float* g2   = (const float*)d_in[9];
    const float* be2  = (const float*)d_in[10];
    const float* W3   = (const float*)d_in[11];
    const float* b3   = (const float*)d_in[12];

    float* out = (float*)d_out;
    float* ws  = (float*)d_ws;
    const int B = in_sizes[0];          // 131072

    float* t2h   = ws;                  // [368]
    float* Lt    = ws + 512;            // [368*384]
    float* logpv = ws + 512 + TPAD * LTW;
    float* xiA   = logpv + B;
    float* hxi2  = xiA + B;
    float* scl   = hxi2 + B;

    param_kernel<<<B / PS, 256, 0, stream>>>(pv, cond, W1, b1, g1, be1,
                                             W2, b2, g2, be2, W3, b3,
                                             logpv, xiA, hxi2, scl);
    chol_kernel<<<1, 512, 0, stream>>>(t2h, Lt);
    gemm_kernel<<<B / ROWS, 256, 0, stream>>>(z, Lt, t2h, logpv, xiA, hxi2, scl, out);
}